// Graph_NN_13271448945380
// MI455X (gfx1250) — compile-verified
//
#include <hip/hip_runtime.h>
#include <hip/hip_bf16.h>

typedef float v2f __attribute__((ext_vector_type(2)));
typedef float v8f __attribute__((ext_vector_type(8)));

#define D 128

// ---------------- utility kernels ----------------

__global__ void zero_u32_kernel(unsigned* __restrict__ p, int n) {
    int i = blockIdx.x * blockDim.x + threadIdx.x;
    if (i < n) p[i] = 0u;
}

__global__ void deg_kernel(const int* __restrict__ dst, unsigned* __restrict__ deg, int E) {
    int e = blockIdx.x * blockDim.x + threadIdx.x;
    if (e < E) atomicAdd(&deg[dst[e]], 1u);
}

__global__ void inv_kernel(const unsigned* __restrict__ deg, float* __restrict__ inv, int n) {
    int i = blockIdx.x * blockDim.x + threadIdx.x;
    if (i < n) inv[i] = 1.0f / ((float)deg[i] + 1.0f);
}

__global__ void embed_kernel(const int* __restrict__ ann, const float* __restrict__ emb,
                             float* __restrict__ h, int total) {
    int i = blockIdx.x * blockDim.x + threadIdx.x;
    if (i < total) {
        int n = i >> 7;       // / D
        int d = i & (D - 1);
        h[i] = emb[(size_t)ann[n] * D + d];
    }
}

// agg[dst] += h[src]; one edge per wave32, each lane moves 4 contiguous floats.
// Gather is a coalesced 512B read per wave; adds are native global_atomic_add_f32
// that resolve in L2 (agg = 51.2MB fits the 192MB L2).
__global__ void scatter_kernel(const float* __restrict__ h, const int* __restrict__ src,
                               const int* __restrict__ dst, float* __restrict__ agg, int E) {
    int tid  = blockIdx.x * blockDim.x + threadIdx.x;
    int e    = tid >> 5;           // wave32: one edge per wave
    if (e >= E) return;
    int lane = tid & 31;
    int s = src[e];
    int d = dst[e];
    const float4 v = *(const float4*)(h + (size_t)s * D + lane * 4);
    float* base = agg + (size_t)d * D + lane * 4;
    unsafeAtomicAdd(base + 0, v.x);
    unsafeAtomicAdd(base + 1, v.y);
    unsafeAtomicAdd(base + 2, v.z);
    unsafeAtomicAdd(base + 3, v.w);
}

// In place: agg = (h + agg) * inv[row]  -> becomes X for the GEMM
__global__ void xform_kernel(const float* __restrict__ h, float* __restrict__ agg,
                             const float* __restrict__ inv, int total) {
    int i = blockIdx.x * blockDim.x + threadIdx.x;
    if (i < total) {
        int n = i >> 7;
        agg[i] = (h[i] + agg[i]) * inv[n];
    }
}

// ---------------- WMMA GEMM: out = relu(X @ W + b) ----------------
// One wave computes a 16x16 output tile. K = 128 fully unrolled as 32
// v_wmma_f32_16x16x4_f32 accumulations (fp32 exact-class, matching reference).
//
// A (16x4 f32) lane layout (ISA 7.12.2): lanes 0-15 = rows M=0..15, VGPR0=K0,
// VGPR1=K1; lanes 16-31 = same rows, VGPR0=K2, VGPR1=K3.
// B (4x16) mirrors: lanes hold column N=lane&15; half-wave selects K pair.
// C/D: VGPR i -> M = i + 8*(lane>=16), N = lane&15.
__global__ __launch_bounds__(256)
void sage_gemm_kernel(const float* __restrict__ X, const float* __restrict__ W,
                      const float* __restrict__ bias, float* __restrict__ out) {
    int tid     = blockIdx.x * blockDim.x + threadIdx.x;
    int wave    = tid >> 5;
    int lane    = tid & 31;
    int rowTile = wave >> 3;       // N/16 row tiles
    int colTile = wave & 7;        // 128/16 = 8 col tiles
    int laneM   = lane & 15;
    int half    = lane >> 4;       // 0: K pair {k,k+1}; 1: {k+2,k+3}
    int rowBase = rowTile * 16;
    int col     = colTile * 16 + laneM;

    const float* arow = X + (size_t)(rowBase + laneM) * D + 2 * half;

    v8f acc = {};
#pragma unroll
    for (int k0 = 0; k0 < D; k0 += 4) {
        v2f a;
        a.x = arow[k0];
        a.y = arow[k0 + 1];
        v2f b;
        b.x = W[(k0 + 2 * half) * D + col];
        b.y = W[(k0 + 2 * half + 1) * D + col];
        acc = __builtin_amdgcn_wmma_f32_16x16x4_f32(
            /*neg_a=*/false, a, /*neg_b=*/false, b,
            /*c_mod=*/(short)0, acc, /*reuse_a=*/false, /*reuse_b=*/false);
    }

    float bv = bias[col];
#pragma unroll
    for (int i = 0; i < 8; ++i) {
        float r = acc[i] + bv;
        out[(size_t)(rowBase + i + 8 * half) * D + col] = r > 0.0f ? r : 0.0f;
    }
}

// ---------------- launcher ----------------

static inline int cdiv(long long a, int b) { return (int)((a + b - 1) / b); }

extern "C" void kernel_launch(void* const* d_in, const int* in_sizes, int n_in,
                              void* d_out, int out_size, void* d_ws, size_t ws_size,
                              hipStream_t stream) {
    const int*   annotation = (const int*)d_in[0];
    const int*   src        = (const int*)d_in[1];
    const int*   dst        = (const int*)d_in[2];
    const float* emb        = (const float*)d_in[3];
    const float* Ws         = (const float*)d_in[4];
    const float* bs         = (const float*)d_in[5];
    float*       out        = (float*)d_out;

    const int N = in_sizes[0];         // 100000 (divisible by 16)
    const int E = in_sizes[1];         // 1600000
    const int L = in_sizes[4] / (D * D);

    // workspace carve: hA[N*D] | agg[N*D] | inv[N] | deg[N]
    float*    hA  = (float*)d_ws;
    float*    agg = hA + (size_t)N * D;
    float*    inv = agg + (size_t)N * D;
    unsigned* deg = (unsigned*)(inv + N);

    const int BT = 256;
    const int totalND = N * D;

    // degrees -> inv (constant across layers)
    zero_u32_kernel<<<cdiv(N, BT), BT, 0, stream>>>(deg, N);
    deg_kernel<<<cdiv(E, BT), BT, 0, stream>>>(dst, deg, E);
    inv_kernel<<<cdiv(N, BT), BT, 0, stream>>>(deg, inv, N);

    // h0 = emb[annotation]
    embed_kernel<<<cdiv(totalND, BT), BT, 0, stream>>>(annotation, emb, hA, totalND);

    // ping-pong: l0: hA->out, l1: out->hA, l2: hA->out (final lands in d_out)
    const float* hin = hA;
    for (int l = 0; l < L; ++l) {
        float* hout = (l & 1) ? hA : out;

        zero_u32_kernel<<<cdiv(totalND, BT), BT, 0, stream>>>((unsigned*)agg, totalND);
        scatter_kernel<<<cdiv((long long)E * 32, BT), BT, 0, stream>>>(hin, src, dst, agg, E);
        xform_kernel<<<cdiv(totalND, BT), BT, 0, stream>>>(hin, agg, inv, totalND);

        int waves = (N / 16) * (D / 16);          // one wave per 16x16 tile
        sage_gemm_kernel<<<cdiv((long long)waves * 32, BT), BT, 0, stream>>>(
            agg, Ws + (size_t)l * D * D, bs + (size_t)l * D, hout);

        hin = hout;
    }
}